// CTCLoss_19593640804729
// MI455X (gfx1250) — compile-verified
//
#include <hip/hip_runtime.h>
#include <math.h>

#define T_ 400
#define B_ 64
#define V_ 5000
#define L_ 50
#define S_ 101          // 2*L+1
#define NEGF (-1e30f)

// ---------------------------------------------------------------------------
// CDNA5 async global->LDS helpers (gfx1250). Per-lane: 16 bytes at gptr are copied to LDS offset lds_off.
// ---------------------------------------------------------------------------
__device__ __forceinline__ void async_load_b128(unsigned lds_off, const void* gptr) {
  asm volatile("global_load_async_to_lds_b128 %0, %1, off"
               :: "v"(lds_off), "v"(gptr) : "memory");
}
#define WAIT_ASYNC(n) asm volatile("s_wait_asynccnt %0" :: "n"(n) : "memory")
#define WAIT_DS0()    asm volatile("s_wait_dscnt 0" ::: "memory")

// ---------------------------------------------------------------------------
// Kernel 1: lse[row] = logsumexp(act[row, 0..V)) for row = t*B + b.
// One block per row; 256 threads; 5 chunks of 256 float4 streamed through a
// 2-deep async LDS double buffer. Each thread keeps its 20 elements in
// registers: pass 1 tracks max only (no expf chain), pass 2 does one
// __expf per element into independent partial sums.
// ---------------------------------------------------------------------------
#define VEC4 (V_ / 4)                                  // 1250 float4 per row
#define K1T  256
#define NCH  ((VEC4 + K1T - 1) / K1T)                  // 5

__global__ __launch_bounds__(K1T) void lse_kernel(const float* __restrict__ act,
                                                  float* __restrict__ lse) {
  const int row = blockIdx.x;
  const int tid = threadIdx.x;
  const float4* gbase = (const float4*)(act + (size_t)row * V_);

  __shared__ float4 stage[2][K1T];
  __shared__ float red_m[8], red_s[8];

  const unsigned loff0 = (unsigned)(size_t)&stage[0][tid];
  const unsigned loff1 = (unsigned)(size_t)&stage[1][tid];

  // Prologue: fill both buffers (chunks 0,1 are always fully in range).
  async_load_b128(loff0, gbase + tid);
  async_load_b128(loff1, gbase + (K1T + tid));

  float4 vals[NCH];
  float m = -3.402823e38f;

#pragma unroll
  for (int c = 0; c < NCH; ++c) {
    if (c + 1 < NCH) { WAIT_ASYNC(1); } else { WAIT_ASYNC(0); }
    const int idx = c * K1T + tid;
    if (idx < VEC4) {
      float4 v = stage[c & 1][tid];
      vals[c] = v;
      m = fmaxf(m, fmaxf(fmaxf(v.x, v.y), fmaxf(v.z, v.w)));
    } else {
      vals[c] = make_float4(-3.402823e38f, -3.402823e38f,
                            -3.402823e38f, -3.402823e38f);
    }
    WAIT_DS0();                       // WAR guard: LDS reads done before reuse
    const int nc = c + 2;
    if (nc < NCH) {
      const int nidx = nc * K1T + tid;
      if (nidx < VEC4) async_load_b128((nc & 1) ? loff1 : loff0, gbase + nidx);
    }
  }

  // One expf per element; 4 independent accumulators (no serial trans chain).
  float s0 = 0.f, s1 = 0.f, s2 = 0.f, s3 = 0.f;
#pragma unroll
  for (int c = 0; c < NCH; ++c) {
    s0 += __expf(vals[c].x - m);
    s1 += __expf(vals[c].y - m);
    s2 += __expf(vals[c].z - m);
    s3 += __expf(vals[c].w - m);
  }
  float s = (s0 + s1) + (s2 + s3);

  // Wave32 reduction of (m, s), then cross-wave via LDS.
#pragma unroll
  for (int off = 16; off > 0; off >>= 1) {
    float m2 = __shfl_xor(m, off, 32);
    float sx = __shfl_xor(s, off, 32);
    float mx = fmaxf(m, m2);
    s = s * __expf(m - mx) + sx * __expf(m2 - mx);
    m = mx;
  }
  const int wave = tid >> 5;
  if ((tid & 31) == 0) { red_m[wave] = m; red_s[wave] = s; }
  __syncthreads();
  if (tid == 0) {
    float M = red_m[0], S = red_s[0];
#pragma unroll
    for (int w = 1; w < 8; ++w) {
      float m2 = red_m[w], sx = red_s[w];
      float mx = fmaxf(M, m2);
      S = S * __expf(M - mx) + sx * __expf(m2 - mx);
      M = mx;
    }
    lse[row] = M + logf(S);
  }
}

// ---------------------------------------------------------------------------
// Kernel 2: CTC alpha recursion. One block per utterance, 128 threads,
// thread s owns extended-label state s (s < 101). Ping-pong alpha in LDS,
// one barrier per time step, depth-8 register prefetch of gathered
// activations act[t, b, ext[s]].
// ---------------------------------------------------------------------------
__global__ __launch_bounds__(128) void alpha_kernel(const float* __restrict__ act,
                                                    const float* __restrict__ lse,
                                                    const int* __restrict__ labels,
                                                    const int* __restrict__ in_len,
                                                    const int* __restrict__ lab_len,
                                                    float* __restrict__ costs) {
  const int b   = blockIdx.x;
  const int tid = threadIdx.x;

  __shared__ float alpha[2][128];
  __shared__ float lse_b[T_];
  __shared__ int   ext_s[S_];

  if (tid < S_) ext_s[tid] = (tid & 1) ? labels[b * L_ + (tid >> 1)] : 0;
  for (int t = tid; t < T_; t += 128) lse_b[t] = lse[t * B_ + b];
  __syncthreads();

  const int Tin = in_len[b];
  const int Ll  = lab_len[b];
  const int e   = (tid < S_) ? ext_s[tid] : 0;
  int skip = 0;
  if (tid < S_ && tid >= 2 && (tid & 1)) skip = (ext_s[tid] != ext_s[tid - 2]) ? 1 : 0;

  const size_t stride = (size_t)B_ * V_;
  const float* col = act + (size_t)b * V_ + e;   // &act[t=0, b, ext[s]]

  // t = 0 init
  {
    float a0 = NEGF;
    if (tid == 0)                 a0 = col[0] - lse_b[0];
    else if (tid == 1 && Ll > 0)  a0 = col[0] - lse_b[0];
    alpha[0][tid] = a0;
  }
  __syncthreads();

  const int CH = 8;
  float buf[CH], nbuf[CH];
#pragma unroll
  for (int k = 0; k < CH; ++k) {
    int t = 1 + k;
    buf[k] = (t < T_ && tid < S_) ? col[(size_t)t * stride] : 0.0f;
  }

  for (int t0 = 1; t0 < T_; t0 += CH) {
    // Prefetch next chunk while this one is consumed.
#pragma unroll
    for (int k = 0; k < CH; ++k) {
      int t = t0 + CH + k;
      nbuf[k] = (t < T_ && tid < S_) ? col[(size_t)t * stride] : 0.0f;
    }
#pragma unroll
    for (int k = 0; k < CH; ++k) {
      const int t = t0 + k;
      if (t < T_) {                       // uniform across block
        const int cur = t & 1, prev = cur ^ 1;
        float a0 = alpha[prev][tid];
        float av;
        if (tid < S_) {
          float a1 = (tid >= 1) ? alpha[prev][tid - 1] : NEGF;
          float a2 = skip ? alpha[prev][tid - 2] : NEGF;
          float mx = fmaxf(a0, fmaxf(a1, a2));
          float ne = mx + logf(expf(a0 - mx) + expf(a1 - mx) + expf(a2 - mx))
                     + (buf[k] - lse_b[t]);
          av = (t < Tin) ? ne : a0;
        } else {
          av = a0;
        }
        alpha[cur][tid] = av;
      }
      __syncthreads();
    }
#pragma unroll
    for (int k = 0; k < CH; ++k) buf[k] = nbuf[k];
  }

  if (tid == 0) {
    const float* aT = alpha[(T_ - 1) & 1];
    float eb = aT[2 * Ll];
    float el = (Ll > 0) ? aT[2 * Ll - 1] : NEGF;
    float mx = fmaxf(eb, el);
    costs[b] = -(mx + logf(expf(eb - mx) + expf(el - mx)));
  }
}

// ---------------------------------------------------------------------------
// Kernel 3: deterministic mean of the 64 per-utterance costs.
// ---------------------------------------------------------------------------
__global__ __launch_bounds__(64) void mean_kernel(const float* __restrict__ costs,
                                                  float* __restrict__ out) {
  const int tid = threadIdx.x;            // 64 threads = 2 waves
  float v = costs[tid];
#pragma unroll
  for (int off = 16; off > 0; off >>= 1) v += __shfl_xor(v, off, 32);
  __shared__ float partial[2];
  if ((tid & 31) == 0) partial[tid >> 5] = v;
  __syncthreads();
  if (tid == 0) out[0] = (partial[0] + partial[1]) * (1.0f / (float)B_);
}

// ---------------------------------------------------------------------------
extern "C" void kernel_launch(void* const* d_in, const int* in_sizes, int n_in,
                              void* d_out, int out_size, void* d_ws, size_t ws_size,
                              hipStream_t stream) {
  const float* act     = (const float*)d_in[0];   // [T, B, V] fp32
  const int*   in_len  = (const int*)d_in[1];     // [B]
  const int*   labels  = (const int*)d_in[2];     // [B, L]
  const int*   lab_len = (const int*)d_in[3];     // [B]

  float* lse   = (float*)d_ws;                    // T*B floats
  float* costs = lse + (size_t)T_ * B_;           // B floats
  float* out   = (float*)d_out;

  lse_kernel  <<<T_ * B_, K1T, 0, stream>>>(act, lse);
  alpha_kernel<<<B_, 128, 0, stream>>>(act, lse, labels, in_len, lab_len, costs);
  mean_kernel <<<1, 64, 0, stream>>>(costs, out);
}